// MambaTemporalEncoder_17214228922833
// MI455X (gfx1250) — compile-verified
//
#include <hip/hip_runtime.h>
#include <hip/hip_bf16.h>
#include <math.h>

#define B_DIM 2
#define L_DIM 1024
#define D_MODEL 512
#define N_LAYERS 4
#define D_STATE 16
#define D_CONV 4
#define D_INNER 1024
#define DT_RANK 32
#define LN_EPS 1e-5f
#define BL (B_DIM * L_DIM)

typedef __attribute__((ext_vector_type(2))) float v2f;
typedef __attribute__((ext_vector_type(8))) float v8f;

// ---------------------------------------------------------------------------
// LayerNorm: one block (256 threads) per row of D floats.
// ---------------------------------------------------------------------------
__global__ void mamba_layernorm_kernel(const float* __restrict__ x,
                                       const float* __restrict__ w,
                                       const float* __restrict__ b,
                                       float* __restrict__ out, int D) {
  const int row = blockIdx.x;
  const float* xr = x + (size_t)row * D;
  float* orow = out + (size_t)row * D;

  __shared__ float red[8];
  const int lane = threadIdx.x & 31;
  const int wid = threadIdx.x >> 5;

  // mean
  float s = 0.f;
  for (int i = threadIdx.x; i < D; i += blockDim.x) s += xr[i];
#pragma unroll
  for (int off = 16; off > 0; off >>= 1) s += __shfl_down(s, off, 32);
  if (lane == 0) red[wid] = s;
  __syncthreads();
  if (threadIdx.x == 0) {
    float tot = 0.f;
#pragma unroll
    for (int i = 0; i < 8; ++i) tot += red[i];
    red[0] = tot / (float)D;
  }
  __syncthreads();
  const float mu = red[0];
  __syncthreads();

  // variance
  float v = 0.f;
  for (int i = threadIdx.x; i < D; i += blockDim.x) {
    float d = xr[i] - mu;
    v += d * d;
  }
#pragma unroll
  for (int off = 16; off > 0; off >>= 1) v += __shfl_down(v, off, 32);
  if (lane == 0) red[wid] = v;
  __syncthreads();
  if (threadIdx.x == 0) {
    float tv = 0.f;
#pragma unroll
    for (int i = 0; i < 8; ++i) tv += red[i];
    red[0] = rsqrtf(tv / (float)D + LN_EPS);
  }
  __syncthreads();
  const float rstd = red[0];

  for (int i = threadIdx.x; i < D; i += blockDim.x)
    orow[i] = (xr[i] - mu) * rstd * w[i] + b[i];
}

// ---------------------------------------------------------------------------
// WMMA fp32 GEMM:  C[M,N] = act( A[M,K(lda)] * W[N,K]^T + bias[N] + res[M,N] )
// One wave per 32x32 tile of C (2x2 grid of 16x16 WMMA tiles; A/B fragments
// each reused twice). K-stepped by 4 via V_WMMA_F32_16X16X4_F32.
// grid = (N/32, M/32), block = 32 threads (one wave, EXEC all ones).
// act: 0 = identity, 1 = softplus.
// ---------------------------------------------------------------------------
__device__ __forceinline__ void gemm_store_tile(float* __restrict__ C,
                                                const float* __restrict__ bias,
                                                const v8f& c, int m0, int n0,
                                                int hl, int l15, int ldc,
                                                int act) {
#pragma unroll
  for (int r = 0; r < 8; ++r) {
    const int m = m0 + r + hl * 8;
    const int n = n0 + l15;
    float v = c[r];
    if (bias != nullptr) v += bias[n];
    if (act == 1) v = (v > 20.f) ? v : log1pf(__expf(v));  // softplus
    C[(size_t)m * ldc + n] = v;
  }
}

__global__ __launch_bounds__(32)
void mamba_gemm_wmma_f32(const float* __restrict__ A,
                         const float* __restrict__ W,
                         const float* __restrict__ bias,
                         const float* __restrict__ res,
                         float* __restrict__ C,
                         int K, int lda, int ldc, int act) {
  const int n0 = blockIdx.x * 32;
  const int m0 = blockIdx.y * 32;
  const int lane = threadIdx.x;
  const int hl = lane >> 4;   // lane half: 0 or 1
  const int l15 = lane & 15;

  v8f c00 = {}, c01 = {}, c10 = {}, c11 = {};
  if (res != nullptr) {
#pragma unroll
    for (int r = 0; r < 8; ++r) {
      const size_t mr0 = (size_t)(m0 + r + hl * 8) * ldc;
      const size_t mr1 = (size_t)(m0 + 16 + r + hl * 8) * ldc;
      c00[r] = res[mr0 + (n0 + l15)];
      c01[r] = res[mr0 + (n0 + 16 + l15)];
      c10[r] = res[mr1 + (n0 + l15)];
      c11[r] = res[mr1 + (n0 + 16 + l15)];
    }
  }

  // A 16x4 (32-bit): lanes 0-15 -> K = k0,k0+1 ; lanes 16-31 -> K = k0+2,k0+3
  const float* arow0 = A + (size_t)(m0 + l15) * lda + hl * 2;
  const float* arow1 = A + (size_t)(m0 + 16 + l15) * lda + hl * 2;
  // B 4x16 = W^T: lanes 0-15 hold K=k0..k0+1, lanes 16-31 hold K=k0+2..k0+3
  const float* wrow0 = W + (size_t)(n0 + l15) * K + hl * 2;
  const float* wrow1 = W + (size_t)(n0 + 16 + l15) * K + hl * 2;

  for (int k0 = 0; k0 < K; k0 += 4) {
    v2f a0, a1, b0, b1;
    a0.x = arow0[k0 + 0]; a0.y = arow0[k0 + 1];
    a1.x = arow1[k0 + 0]; a1.y = arow1[k0 + 1];
    b0.x = wrow0[k0 + 0]; b0.y = wrow0[k0 + 1];
    b1.x = wrow1[k0 + 0]; b1.y = wrow1[k0 + 1];
    c00 = __builtin_amdgcn_wmma_f32_16x16x4_f32(false, a0, false, b0,
                                                (short)0, c00, false, false);
    c01 = __builtin_amdgcn_wmma_f32_16x16x4_f32(false, a0, false, b1,
                                                (short)0, c01, false, false);
    c10 = __builtin_amdgcn_wmma_f32_16x16x4_f32(false, a1, false, b0,
                                                (short)0, c10, false, false);
    c11 = __builtin_amdgcn_wmma_f32_16x16x4_f32(false, a1, false, b1,
                                                (short)0, c11, false, false);
  }

  gemm_store_tile(C, bias, c00, m0,      n0,      hl, l15, ldc, act);
  gemm_store_tile(C, bias, c01, m0,      n0 + 16, hl, l15, ldc, act);
  gemm_store_tile(C, bias, c10, m0 + 16, n0,      hl, l15, ldc, act);
  gemm_store_tile(C, bias, c11, m0 + 16, n0 + 16, hl, l15, ldc, act);
}

// ---------------------------------------------------------------------------
// Causal depthwise conv (width 4) over time + SiLU.
// xz layout (B, L, 2*D_INNER); u-half is columns [0, D_INNER).
// ---------------------------------------------------------------------------
__global__ void mamba_conv_silu_kernel(const float* __restrict__ xz,
                                       const float* __restrict__ cw,
                                       const float* __restrict__ cb,
                                       float* __restrict__ u) {
  const size_t idx = (size_t)blockIdx.x * blockDim.x + threadIdx.x;
  if (idx >= (size_t)BL * D_INNER) return;
  const int d = (int)(idx % D_INNER);
  const size_t bl = idx / D_INNER;
  const int l = (int)(bl % L_DIM);
  const int b = (int)(bl / L_DIM);

  const float* base = xz + ((size_t)b * L_DIM) * (2 * D_INNER) + d;
  float acc = cb[d];
#pragma unroll
  for (int j = 0; j < D_CONV; ++j) {
    const int t = l - (D_CONV - 1) + j;
    if (t >= 0) acc += cw[d * D_CONV + j] * base[(size_t)t * (2 * D_INNER)];
  }
  const float sg = 1.f / (1.f + __expf(-acc));
  u[idx] = acc * sg;  // silu
}

// ---------------------------------------------------------------------------
// Selective scan: one thread per (b, d); h[16] and A-row in registers.
// B_t / C_t (32 floats) double-buffered in LDS via GLOBAL_LOAD_ASYNC_TO_LDS:
// wave 0 prefetches step t+1 while the whole block computes step t.
// grid = B * (D_INNER/256), block = 256.
// ---------------------------------------------------------------------------
__global__ void mamba_scan_kernel(const float* __restrict__ dt,
                                  const float* __restrict__ u,
                                  const float* __restrict__ xdbl,
                                  const float* __restrict__ A_log,
                                  const float* __restrict__ Dp,
                                  float* __restrict__ y) {
  const int blocks_per_b = D_INNER / 256;
  const int b = blockIdx.x / blocks_per_b;
  const int d = (blockIdx.x % blocks_per_b) * 256 + threadIdx.x;

  float Arow[D_STATE], h[D_STATE];
#pragma unroll
  for (int n = 0; n < D_STATE; ++n) {
    Arow[n] = -__expf(A_log[(size_t)d * D_STATE + n]);
    h[n] = 0.f;
  }
  const float dskip = Dp[d];

  __shared__ float BC[2][2 * D_STATE];  // double buffer, 32 floats each
  const float* gsrc = xdbl + ((size_t)b * L_DIM) * 64 + DT_RANK;  // B@+0, C@+16
  const float* dtp = dt + ((size_t)b * L_DIM) * D_INNER + d;
  const float* up = u + ((size_t)b * L_DIM) * D_INNER + d;
  float* yp = y + ((size_t)b * L_DIM) * D_INNER + d;

  // Prefetch t = 0 into buffer 0 (wave 0 only; 32 lanes, one float each).
  if (threadIdx.x < 32) {
    const float* g0 = gsrc + threadIdx.x;
    const unsigned int l0 =
        (unsigned int)(uintptr_t)&BC[0][threadIdx.x];  // LDS byte offset
    asm volatile("global_load_async_to_lds_b32 %0, %1, off"
                 :: "v"(l0), "v"(g0) : "memory");
  }

  for (int t = 0; t < L_DIM; ++t) {
    const int cur = t & 1;
    if (threadIdx.x < 32) {
      if (t + 1 < L_DIM) {
        // issue prefetch of step t+1 into the other buffer, then wait until
        // only that newest request is outstanding (step t's data has landed)
        const float* gn = gsrc + (size_t)(t + 1) * 64 + threadIdx.x;
        const unsigned int ln =
            (unsigned int)(uintptr_t)&BC[(t + 1) & 1][threadIdx.x];
        asm volatile("global_load_async_to_lds_b32 %0, %1, off"
                     :: "v"(ln), "v"(gn) : "memory");
        asm volatile("s_wait_asynccnt 0x1" ::: "memory");
      } else {
        asm volatile("s_wait_asynccnt 0x0" ::: "memory");
      }
    }
    __syncthreads();

    const float dtv = dtp[(size_t)t * D_INNER];
    const float uv = up[(size_t)t * D_INNER];
    const float du = dtv * uv;
    float yv = 0.f;
#pragma unroll
    for (int n = 0; n < D_STATE; ++n) {
      h[n] = __expf(dtv * Arow[n]) * h[n] + du * BC[cur][n];
      yv += h[n] * BC[cur][D_STATE + n];
    }
    yp[(size_t)t * D_INNER] = yv + uv * dskip;
    __syncthreads();
  }
}

// ---------------------------------------------------------------------------
// In-place gating: y *= silu(z), z = xz[..., D_INNER + d]
// ---------------------------------------------------------------------------
__global__ void mamba_gate_kernel(float* __restrict__ y,
                                  const float* __restrict__ xz) {
  const size_t idx = (size_t)blockIdx.x * blockDim.x + threadIdx.x;
  if (idx >= (size_t)BL * D_INNER) return;
  const int d = (int)(idx % D_INNER);
  const size_t bl = idx / D_INNER;
  const float z = xz[bl * (2 * D_INNER) + D_INNER + d];
  const float sg = z / (1.f + __expf(-z));
  y[idx] *= sg;
}

// ---------------------------------------------------------------------------
extern "C" void kernel_launch(void* const* d_in, const int* in_sizes, int n_in,
                              void* d_out, int out_size, void* d_ws, size_t ws_size,
                              hipStream_t stream) {
  const float* x_in   = (const float*)d_in[0];
  const float* norm_w = (const float*)d_in[1];
  const float* norm_b = (const float*)d_in[2];
  const float* in_w   = (const float*)d_in[3];   // (4, 2048, 512)
  const float* conv_w = (const float*)d_in[4];   // (4, 1024, 4)
  const float* conv_b = (const float*)d_in[5];   // (4, 1024)
  const float* xp_w   = (const float*)d_in[6];   // (4, 64, 1024)
  const float* dt_w   = (const float*)d_in[7];   // (4, 1024, 32)
  const float* dt_b   = (const float*)d_in[8];   // (4, 1024)
  const float* A_log  = (const float*)d_in[9];   // (4, 1024, 16)
  const float* D_skip = (const float*)d_in[10];  // (4, 1024)
  const float* out_w  = (const float*)d_in[11];  // (4, 512, 1024)
  const float* fin_w  = (const float*)d_in[12];
  const float* fin_b  = (const float*)d_in[13];

  // Workspace carve-up (floats)
  float* ws = (float*)d_ws;
  size_t o = 0;
  float* xbuf0 = ws + o; o += (size_t)BL * D_MODEL;        // 1M
  float* xbuf1 = ws + o; o += (size_t)BL * D_MODEL;        // 1M
  float* xln   = ws + o; o += (size_t)BL * D_MODEL;        // 1M
  float* xz    = ws + o; o += (size_t)BL * 2 * D_INNER;    // 4M
  float* u     = ws + o; o += (size_t)BL * D_INNER;        // 2M
  float* xdbl  = ws + o; o += (size_t)BL * 64;             // 128K
  float* dtbuf = ws + o; o += (size_t)BL * D_INNER;        // 2M
  float* ybuf  = ws + o; o += (size_t)BL * D_INNER;        // 2M

  hipMemcpyAsync(xbuf0, x_in, (size_t)BL * D_MODEL * sizeof(float),
                 hipMemcpyDeviceToDevice, stream);

  float* xc = xbuf0;
  float* xn = xbuf1;

  const size_t n_elem = (size_t)BL * D_INNER;
  const int ew_blocks = (int)((n_elem + 255) / 256);

  for (int layer = 0; layer < N_LAYERS; ++layer) {
    // 1) pre-norm
    mamba_layernorm_kernel<<<BL, 256, 0, stream>>>(
        xc, norm_w + (size_t)layer * D_MODEL, norm_b + (size_t)layer * D_MODEL,
        xln, D_MODEL);

    // 2) in_proj: xz = xln (BLx512) * in_w^T (512x2048)
    mamba_gemm_wmma_f32<<<dim3(2 * D_INNER / 32, BL / 32), 32, 0, stream>>>(
        xln, in_w + (size_t)layer * 2 * D_INNER * D_MODEL,
        nullptr, nullptr, xz, D_MODEL, D_MODEL, 2 * D_INNER, 0);

    // 3) causal depthwise conv + silu -> u
    mamba_conv_silu_kernel<<<ew_blocks, 256, 0, stream>>>(
        xz, conv_w + (size_t)layer * D_INNER * D_CONV,
        conv_b + (size_t)layer * D_INNER, u);

    // 4) x_proj: xdbl = u (BLx1024) * xp_w^T (1024x64)
    mamba_gemm_wmma_f32<<<dim3(64 / 32, BL / 32), 32, 0, stream>>>(
        u, xp_w + (size_t)layer * 64 * D_INNER,
        nullptr, nullptr, xdbl, D_INNER, D_INNER, 64, 0);

    // 5) dt_proj + softplus: dt = softplus(xdbl[:, :32] * dt_w^T + dt_b)
    mamba_gemm_wmma_f32<<<dim3(D_INNER / 32, BL / 32), 32, 0, stream>>>(
        xdbl, dt_w + (size_t)layer * D_INNER * DT_RANK,
        dt_b + (size_t)layer * D_INNER, nullptr, dtbuf,
        DT_RANK, 64, D_INNER, 1);

    // 6) selective scan -> ybuf (includes + u * D_skip)
    mamba_scan_kernel<<<B_DIM * (D_INNER / 256), 256, 0, stream>>>(
        dtbuf, u, xdbl, A_log + (size_t)layer * D_INNER * D_STATE,
        D_skip + (size_t)layer * D_INNER, ybuf);

    // 7) gating: ybuf *= silu(z)
    mamba_gate_kernel<<<ew_blocks, 256, 0, stream>>>(ybuf, xz);

    // 8) out_proj + residual: xn = ybuf (BLx1024) * out_w^T (1024x512) + xc
    mamba_gemm_wmma_f32<<<dim3(D_MODEL / 32, BL / 32), 32, 0, stream>>>(
        ybuf, out_w + (size_t)layer * D_MODEL * D_INNER,
        nullptr, xc, xn, D_INNER, D_INNER, D_MODEL, 0);

    // swap buffers
    float* tmp = xc; xc = xn; xn = tmp;
  }

  // final layernorm into d_out
  mamba_layernorm_kernel<<<BL, 256, 0, stream>>>(xc, fin_w, fin_b,
                                                 (float*)d_out, D_MODEL);
}